// LinearEmbedding_8418135900659
// MI455X (gfx1250) — compile-verified
//
#include <hip/hip_runtime.h>

// out[s, :] = W_tok[x[s], :] + b_tok + W_pos[(S-1)-s, :] + b_pos
// S = 8192, D = 1024, fp32.  Pure HBM-bound gather+add: ~96 MB of traffic,
// ~25 MFLOP -> 0.26 FLOP/B.  No WMMA by design (dense one-hot matmul would be
// ~0.85 PFLOP + 1.6 GB one-hot, strictly worse).  Target: saturate 23.3 TB/s
// with b128 vector memory ops; stage the random W_tok gather through the
// gfx1250 async-to-LDS path (ASYNCcnt + s_wait_asynccnt).

typedef float v4f __attribute__((ext_vector_type(4)));
typedef int   v4i __attribute__((vector_size(16)));   // matches builtin param

#define EMBED_DIM       1024
#define VEC_PER_ROW     (EMBED_DIM / 4)   // 256 float4 per row
#define THREADS         VEC_PER_ROW       // one float4 per lane per row
#define ROWS_PER_BLOCK  4

#if defined(__HIP_DEVICE_COMPILE__) && defined(__gfx1250__) && \
    __has_builtin(__builtin_amdgcn_global_load_async_to_lds_b128)
#define USE_ASYNC_LDS 1
#else
#define USE_ASYNC_LDS 0
#endif

#if USE_ASYNC_LDS
typedef __attribute__((address_space(1))) v4i* as1_v4i;   // global 128b
typedef __attribute__((address_space(3))) v4i* as3_v4i;   // LDS 128b

__device__ __forceinline__ void wait_async0() {
#if __has_builtin(__builtin_amdgcn_s_wait_asynccnt)
  __builtin_amdgcn_s_wait_asynccnt(0);
#else
  asm volatile("s_wait_asynccnt 0" ::: "memory");
#endif
}
#endif

__global__ __launch_bounds__(THREADS)
void embed_sum_kernel(const int*   __restrict__ x,
                      const float* __restrict__ W_tok,
                      const float* __restrict__ b_tok,
                      const float* __restrict__ W_pos,
                      const float* __restrict__ b_pos,
                      float*       __restrict__ out,
                      int S)
{
  const int c4   = threadIdx.x;                    // float4 slot within a row
  const int row0 = blockIdx.x * ROWS_PER_BLOCK;

  // Biases: 8 KB total, hot in cache; fold into one register vector.
  const v4f* bt = reinterpret_cast<const v4f*>(b_tok);
  const v4f* bp = reinterpret_cast<const v4f*>(b_pos);
  const v4f  vb = bt[c4] + bp[c4];

#if USE_ASYNC_LDS
  // Stage the random-gather rows of W_tok through LDS with the async DMA path.
  // Each lane async-copies exactly the 16B slice it will later read, so only
  // an ASYNCcnt wait is required (no workgroup barrier).  All 4 rows are
  // issued before the first wait to overlap their gather latency.
  __shared__ v4f tok_tile[ROWS_PER_BLOCK][VEC_PER_ROW];
  #pragma unroll
  for (int i = 0; i < ROWS_PER_BLOCK; ++i) {
    const int row = row0 + i;
    if (row < S) {
      const int tok = x[row];
      const v4f* wt = reinterpret_cast<const v4f*>(W_tok)
                    + (size_t)tok * VEC_PER_ROW;
      __builtin_amdgcn_global_load_async_to_lds_b128(
          (as1_v4i)(wt + c4), (as3_v4i)&tok_tile[i][c4],
          /*offset=*/0, /*cpol=*/0);
    }
  }
#endif

  #pragma unroll
  for (int i = 0; i < ROWS_PER_BLOCK; ++i) {
    const int row = row0 + i;
    if (row >= S) break;
    const int pos = (S - 1) - row;

    // W_pos rows are streamed exactly once -> non-temporal b128 load.
    const v4f* wp = reinterpret_cast<const v4f*>(W_pos)
                  + (size_t)pos * VEC_PER_ROW;
    v4f vp = __builtin_nontemporal_load(wp + c4);

#if USE_ASYNC_LDS
    if (i == 0) wait_async0();          // all 4 async gathers complete
    v4f vt = tok_tile[i][c4];
#else
    const int tok = x[row];
    const v4f* wt = reinterpret_cast<const v4f*>(W_tok)
                  + (size_t)tok * VEC_PER_ROW;
    v4f vt = wt[c4];
#endif

    v4f r = vt + vp + vb;

    // Output written once, never re-read -> non-temporal b128 store.
    v4f* o = reinterpret_cast<v4f*>(out) + (size_t)row * VEC_PER_ROW + c4;
    __builtin_nontemporal_store(r, o);
  }
}

extern "C" void kernel_launch(void* const* d_in, const int* in_sizes, int n_in,
                              void* d_out, int out_size, void* d_ws, size_t ws_size,
                              hipStream_t stream) {
  (void)n_in; (void)out_size; (void)d_ws; (void)ws_size;
  const int    S     = in_sizes[0];                 // 8192
  const int*   x     = (const int*)  d_in[0];
  const float* W_tok = (const float*)d_in[1];
  const float* b_tok = (const float*)d_in[2];
  const float* W_pos = (const float*)d_in[3];
  const float* b_pos = (const float*)d_in[4];
  float*       out   = (float*)d_out;

  const int nblocks = (S + ROWS_PER_BLOCK - 1) / ROWS_PER_BLOCK;
  embed_sum_kernel<<<dim3(nblocks), dim3(THREADS), 0, stream>>>(
      x, W_tok, b_tok, W_pos, b_pos, out, S);
}